// BallPredictorGNN_70334384439314
// MI455X (gfx1250) — compile-verified
//
#include <hip/hip_runtime.h>
#include <math.h>

typedef __attribute__((ext_vector_type(2))) float v2f;
typedef __attribute__((ext_vector_type(8))) float v8f;
typedef __attribute__((ext_vector_type(4))) unsigned int u32x4;
typedef __attribute__((ext_vector_type(8))) int i32x8;
typedef __attribute__((ext_vector_type(4))) int i32x4;

#define LRELU(v) ((v) >= 0.0f ? (v) : 0.2f * (v))

__device__ __forceinline__ void atomicMaxF(float* addr, float val) {
  // monotone float ordering via int (non-neg) / uint (neg) atomics; init must be -inf
  if (val >= 0.0f) atomicMax((int*)addr, __float_as_int(val));
  else             atomicMin((unsigned int*)addr, __float_as_uint(val));
}

// ---------------- fill ----------------
__global__ void fill_f32(float* __restrict__ p, int n, float v) {
  int i = blockIdx.x * blockDim.x + threadIdx.x;
  if (i < n) p[i] = v;
}

// ---------------- WMMA f32 GEMM: D[M,Nc] = A[M,K] @ B[K,Nc], row-major ----------------
// Block = 4 waves covering one 16-row M tile x four 16-col N tiles (ntiles % 4 == 0).
// A tile (16 x K) is staged into LDS once per block via the Tensor Data Mover.
__global__ void gemm_wmma_f32(const float* __restrict__ A, const float* __restrict__ B,
                              float* __restrict__ D, int M, int K, int Nc) {
  __shared__ float lds_a[16 * 256];                 // K <= 256
  const int lane = threadIdx.x & 31;
  const int wid  = threadIdx.x >> 5;
  const int ntiles = Nc >> 4;
  const int tile0 = blockIdx.x * 4;                 // ntiles % 4 == 0 => same M tile
  const int mt = tile0 / ntiles;
  const int m0 = mt << 4;
  const int n0 = ((tile0 - mt * ntiles) + wid) << 4;

  if (wid == 0) {
    // TDM: 2D tile load, 16 rows x K floats, row stride K, into LDS (contiguous).
    unsigned long long ga = (unsigned long long)(const void*)(A + (long)m0 * K);
    unsigned int ldsoff = (unsigned int)(size_t)(void*)lds_a;  // low 32 bits = LDS byte addr
    u32x4 g0;
    g0[0] = 1u;                                            // count=1, user descriptor
    g0[1] = ldsoff;                                        // lds_addr (bytes)
    g0[2] = (unsigned int)(ga & 0xFFFFFFFFu);              // global_addr[31:0]
    g0[3] = (unsigned int)((ga >> 32) & 0x01FFFFFFu) | (2u << 30);  // addr[56:32] | type=2
    i32x8 g1;
    g1[0] = (int)(2u << 16);                               // wg_mask=0, data_size=4B
    g1[1] = (int)(((unsigned)K & 0xFFFFu) << 16);          // tensor_dim0[15:0]=K
    g1[2] = (int)(((unsigned)K >> 16) | (16u << 16));      // tensor_dim0[31:16] | tensor_dim1=16
    g1[3] = (int)(((unsigned)K & 0xFFFFu) << 16);          // tensor_dim1[31:16]=0 | tile_dim0=K
    g1[4] = 16;                                            // tile_dim1=16, tile_dim2=0
    g1[5] = K;                                             // tensor_dim0_stride[31:0]=K
    g1[6] = 0;                                             // stride0[47:32]=0, stride1[15:0]=0
    g1[7] = 0;                                             // stride1[47:16]=0
    i32x4 gz4 = {0, 0, 0, 0};                              // 2D: groups 2/3 unused
    i32x8 gz8 = {0, 0, 0, 0, 0, 0, 0, 0};
    __builtin_amdgcn_tensor_load_to_lds(g0, g1, gz4, gz4, gz8, 0);
    __builtin_amdgcn_s_wait_tensorcnt(0);
  }
  __syncthreads();

  const int r  = lane & 15;
  const int kq = (lane >> 4) << 1;                  // 0 (lanes 0-15) / 2 (lanes 16-31)
  const float* la = lds_a + r * K + kq;             // A 16x4 operand: vgpr0/1 = K(kq),K(kq+1)
  const float* pb = B + (long)kq * Nc + n0 + r;     // B 4x16 operand: rows kq, kq+1
  const long bstep = 4L * (long)Nc;
  v8f acc0 = {}, acc1 = {};
  for (int k0 = 0; k0 < K; k0 += 8) {               // two independent accumulator chains
    v2f a0, b0, a1, b1;
    a0.x = la[0]; a0.y = la[1];
    b0.x = pb[0]; b0.y = pb[Nc];
    a1.x = la[4]; a1.y = la[5];
    b1.x = pb[bstep]; b1.y = pb[bstep + Nc];
    __builtin_prefetch(pb + 2 * bstep, 0, 1);       // global_prefetch_b8
    acc0 = __builtin_amdgcn_wmma_f32_16x16x4_f32(false, a0, false, b0, (short)0, acc0,
                                                 false, false);
    acc1 = __builtin_amdgcn_wmma_f32_16x16x4_f32(false, a1, false, b1, (short)0, acc1,
                                                 false, false);
    la += 8;
    pb += 2 * bstep;
  }
  // D: VGPR v -> row m0 + (lane>=16 ? 8 : 0) + v, col n0 + (lane&15)
  float* Drow = D + (long)(m0 + (lane >> 4) * 8) * Nc + n0 + r;
#pragma unroll
  for (int v = 0; v < 8; ++v) Drow[(long)v * Nc] = acc0[v] + acc1[v];
}

// ---------------- per-(node,head) attention logits, C=64, one wave each ----------------
__global__ void attn_logits(const float* __restrict__ h, const float* __restrict__ wsrc,
                            const float* __restrict__ wdst, float* __restrict__ as,
                            float* __restrict__ ad, int N, int H) {
  int gw = (blockIdx.x * blockDim.x + threadIdx.x) >> 5;
  int lane = threadIdx.x & 31;
  if (gw >= N * H) return;
  int n = gw / H, hh = gw - n * H;
  const float* hp = h + ((long)n * H + hh) * 64;
  const float* sp = wsrc + hh * 64;
  const float* dp = wdst + hh * 64;
  float ss = hp[lane] * sp[lane] + hp[lane + 32] * sp[lane + 32];
  float sd = hp[lane] * dp[lane] + hp[lane + 32] * dp[lane + 32];
  for (int off = 16; off; off >>= 1) {
    ss += __shfl_xor(ss, off);
    sd += __shfl_xor(sd, off);
  }
  if (lane == 0) { as[(long)n * H + hh] = ss; ad[(long)n * H + hh] = sd; }
}

// ---------------- mark sources of edges into the ball node ----------------
__global__ void mark_sources(const long long* __restrict__ ei, int E, int N,
                             int* __restrict__ bmp) {
  int e = blockIdx.x * blockDim.x + threadIdx.x;
  if (e == 0) bmp[N - 1] = 1;                    // ball itself (self-loop source)
  if (e >= E) return;
  if ((int)ei[E + e] == N - 1) bmp[(int)ei[e]] = 1;
}

// ---------------- edge softmax pass A: running max per (dst, head) ----------------
__global__ void edge_max(const long long* __restrict__ ei, int E, int N, int H,
                         const float* __restrict__ as, const float* __restrict__ ad,
                         float* __restrict__ m, int only_dst) {
  int tid = blockIdx.x * blockDim.x + threadIdx.x;
  if (tid >= (E + N) * H) return;
  int e = tid / H, hh = tid - (tid / H) * H;
  int src, dst;
  if (e < E) { src = (int)ei[e]; dst = (int)ei[E + e]; } else { src = dst = e - E; }
  if (only_dst >= 0 && dst != only_dst) return;
  float v = as[(long)src * H + hh] + ad[(long)dst * H + hh];
  v = LRELU(v);
  atomicMaxF(&m[(long)dst * H + hh], v);
}

// ---------------- edge softmax pass B: sum of exp(e - max) per (dst, head) ----------------
__global__ void edge_sum(const long long* __restrict__ ei, int E, int N, int H,
                         const float* __restrict__ as, const float* __restrict__ ad,
                         const float* __restrict__ m, float* __restrict__ s, int only_dst) {
  int tid = blockIdx.x * blockDim.x + threadIdx.x;
  if (tid >= (E + N) * H) return;
  int e = tid / H, hh = tid - (tid / H) * H;
  int src, dst;
  if (e < E) { src = (int)ei[e]; dst = (int)ei[E + e]; } else { src = dst = e - E; }
  if (only_dst >= 0 && dst != only_dst) return;
  float v = as[(long)src * H + hh] + ad[(long)dst * H + hh];
  v = LRELU(v);
  atomicAdd(&s[(long)dst * H + hh], expf(v - m[(long)dst * H + hh]));
}

// ---------------- layer-1 aggregation (H=4, C=64), restricted to bitmap[dst] ----------------
__global__ void aggregate_h4(const long long* __restrict__ ei, int E, int N,
                             const float* __restrict__ h, const float* __restrict__ as,
                             const float* __restrict__ ad, const float* __restrict__ m,
                             const float* __restrict__ s, const int* __restrict__ bmp,
                             float* __restrict__ out) {
  int gw = (blockIdx.x * blockDim.x + threadIdx.x) >> 5;   // wave per edge
  int lane = threadIdx.x & 31;
  if (gw >= E + N) return;
  int src, dst;
  if (gw < E) { src = (int)ei[gw]; dst = (int)ei[E + gw]; } else { src = dst = gw - E; }
  if (!bmp[dst]) return;                                   // wave-uniform
  int hh = lane & 3;                                       // lanes 0..3 hold heads 0..3
  float v = as[(long)src * 4 + hh] + ad[(long)dst * 4 + hh];
  v = LRELU(v);
  float alpha = expf(v - m[(long)dst * 4 + hh]) / (s[(long)dst * 4 + hh] + 1e-16f);
  const float* hp = h + (long)src * 256;
  float* op = out + (long)dst * 256;
#pragma unroll
  for (int i = 0; i < 8; ++i) {
    int idx = lane + (i << 5);
    float a = __shfl(alpha, idx >> 6);                     // broadcast head idx/64
    atomicAdd(op + idx, hp[idx] * a);
  }
}

// ---------------- layer-2 aggregation: only dst == ball (H=1, C=64) ----------------
__global__ void aggregate_ball(const long long* __restrict__ ei, int E, int N,
                               const float* __restrict__ h2, const float* __restrict__ as,
                               const float* __restrict__ ad, const float* __restrict__ m,
                               const float* __restrict__ s, float* __restrict__ out64) {
  int e = blockIdx.x * blockDim.x + threadIdx.x;
  if (e >= E + N) return;
  int src, dst;
  if (e < E) { src = (int)ei[e]; dst = (int)ei[E + e]; } else { src = dst = e - E; }
  int ball = N - 1;
  if (dst != ball) return;
  float v = as[src] + ad[ball];
  v = LRELU(v);
  float alpha = expf(v - m[ball]) / (s[ball] + 1e-16f);
  const float* hp = h2 + (long)src * 64;
  for (int c = 0; c < 64; ++c) atomicAdd(out64 + c, hp[c] * alpha);
}

// ---------------- bias + relu (in place) ----------------
__global__ void bias_relu(float* __restrict__ x, const float* __restrict__ b, int n,
                          int stride) {
  int i = blockIdx.x * blockDim.x + threadIdx.x;
  if (i >= n) return;
  float v = x[i] + b[i % stride];
  x[i] = v > 0.0f ? v : 0.0f;
}

// ---------------- final MLP on ball node: 64 -> 32 -> 2 ----------------
__global__ void head_mlp(const float* __restrict__ agg2, const float* __restrict__ b2,
                         const float* __restrict__ fc1w, const float* __restrict__ fc1b,
                         const float* __restrict__ fc2w, const float* __restrict__ fc2b,
                         float* __restrict__ out) {
  __shared__ float ball[64];
  __shared__ float z[32];
  int t = threadIdx.x;
  float v = agg2[t] + b2[t];
  ball[t] = v > 0.0f ? v : 0.0f;
  __syncthreads();
  if (t < 32) {
    float acc = fc1b[t];
    for (int c = 0; c < 64; ++c) acc += ball[c] * fc1w[c * 32 + t];
    z[t] = acc > 0.0f ? acc : 0.0f;
  }
  __syncthreads();
  if (t < 2) {
    float acc = fc2b[t];
    for (int j = 0; j < 32; ++j) acc += z[j] * fc2w[j * 2 + t];
    out[t] = acc;
  }
}

extern "C" void kernel_launch(void* const* d_in, const int* in_sizes, int n_in,
                              void* d_out, int out_size, void* d_ws, size_t ws_size,
                              hipStream_t stream) {
  const float*     x    = (const float*)d_in[0];
  const long long* ei   = (const long long*)d_in[1];   // int64 per reference
  const float*     W1   = (const float*)d_in[2];
  const float*     as1w = (const float*)d_in[3];
  const float*     ad1w = (const float*)d_in[4];
  const float*     b1   = (const float*)d_in[5];
  const float*     W2   = (const float*)d_in[6];
  const float*     as2w = (const float*)d_in[7];
  const float*     ad2w = (const float*)d_in[8];
  const float*     b2   = (const float*)d_in[9];
  const float*     fc1w = (const float*)d_in[10];
  const float*     fc1b = (const float*)d_in[11];
  const float*     fc2w = (const float*)d_in[12];
  const float*     fc2b = (const float*)d_in[13];
  float*           out  = (float*)d_out;

  const int N  = in_sizes[0] / 128;   // F_in = 128
  const int E  = in_sizes[1] / 2;
  const int ET = E + N;

  // workspace layout (floats)
  float* p    = (float*)d_ws;
  float* h1   = p; p += (long)N * 256;   // layer1 features [N,4,64]
  float* agg1 = p; p += (long)N * 256;   // layer1 aggregate -> hr1 in place
  float* h2   = p; p += (long)N * 64;    // layer2 features [N,64]
  float* as1  = p; p += (long)N * 4;
  float* ad1  = p; p += (long)N * 4;
  float* m1   = p; p += (long)N * 4;
  float* s1   = p; p += (long)N * 4;
  float* as2  = p; p += N;
  float* ad2  = p; p += N;
  float* m2   = p; p += N;
  float* s2   = p; p += N;
  int*   bmp  = (int*)p; p += N;
  float* agg2 = p; p += 64;

  // per-call init (graph-capture safe)
  (void)hipMemsetAsync(agg1, 0, (size_t)N * 256 * sizeof(float), stream);
  (void)hipMemsetAsync(s1,   0, (size_t)N * 4 * sizeof(float), stream);
  (void)hipMemsetAsync(s2,   0, (size_t)N * sizeof(float), stream);
  (void)hipMemsetAsync(bmp,  0, (size_t)N * sizeof(int), stream);
  (void)hipMemsetAsync(agg2, 0, 64 * sizeof(float), stream);
  fill_f32<<<(N * 4 + 255) / 256, 256, 0, stream>>>(m1, N * 4, -INFINITY);
  fill_f32<<<(N + 255) / 256, 256, 0, stream>>>(m2, N, -INFINITY);

  // layer 1: h1 = x @ W1  (WMMA f32 + TDM-staged A tile)
  {
    int tiles = (N / 16) * (256 / 16);
    gemm_wmma_f32<<<tiles / 4, 128, 0, stream>>>(x, W1, h1, N, 128, 256);
  }
  attn_logits<<<((long)N * 4 * 32 + 255) / 256, 256, 0, stream>>>(h1, as1w, ad1w, as1, ad1, N, 4);
  mark_sources<<<(E + 255) / 256, 256, 0, stream>>>(ei, E, N, bmp);
  edge_max<<<((long)ET * 4 + 255) / 256, 256, 0, stream>>>(ei, E, N, 4, as1, ad1, m1, -1);
  edge_sum<<<((long)ET * 4 + 255) / 256, 256, 0, stream>>>(ei, E, N, 4, as1, ad1, m1, s1, -1);
  aggregate_h4<<<((long)ET * 32 + 255) / 256, 256, 0, stream>>>(ei, E, N, h1, as1, ad1, m1, s1, bmp, agg1);
  bias_relu<<<(N * 256 + 255) / 256, 256, 0, stream>>>(agg1, b1, N * 256, 256);

  // layer 2: h2 = hr1 @ W2  (WMMA f32 + TDM-staged A tile)
  {
    int tiles = (N / 16) * (64 / 16);
    gemm_wmma_f32<<<tiles / 4, 128, 0, stream>>>(agg1, W2, h2, N, 256, 64);
  }
  attn_logits<<<((long)N * 32 + 255) / 256, 256, 0, stream>>>(h2, as2w, ad2w, as2, ad2, N, 1);
  int ball = N - 1;
  edge_max<<<(ET + 255) / 256, 256, 0, stream>>>(ei, E, N, 1, as2, ad2, m2, ball);
  edge_sum<<<(ET + 255) / 256, 256, 0, stream>>>(ei, E, N, 1, as2, ad2, m2, s2, ball);
  aggregate_ball<<<(ET + 255) / 256, 256, 0, stream>>>(ei, E, N, h2, as2, ad2, m2, s2, agg2);

  head_mlp<<<1, 64, 0, stream>>>(agg2, b2, fc1w, fc1b, fc2w, fc2b, out);
}